// InverseArnoldCat_23536420782185
// MI455X (gfx1250) — compile-verified
//
#include <hip/hip_runtime.h>

// Inverse Arnold cat map, 7 iterations fused into one gather via T^7 mod 512.
// T = [[1,-1],[-2,1]]  =>  T^7 ≡ [[239,343],[174,239]] (mod 512)
// out[b,i,j,ch] = in[b, (239*i+343*j)&511, (174*i+239*j)&511, ch]
//
// Pure data movement: 192 MiB read + 192 MiB write -> ~16.5 us at 23.3 TB/s.
// Each 3 MiB batch image is gathered by its own 128 blocks, so scattered 12 B
// reads are L2-resident; output uses CDNA5 non-temporal stores (th:TH_STORE_NT)
// so the write stream does not evict the gather working set.

#define PPT 8  // pixels per thread; one block covers 2048 consecutive pixels

__device__ __forceinline__ unsigned src_index(unsigned p) {
  const unsigned j = p & 511u;
  const unsigned i = (p >> 9) & 511u;
  const unsigned b = p >> 18;
  const unsigned r = (239u * i + 343u * j) & 511u;
  const unsigned c = (174u * i + 239u * j) & 511u;
  return (((b << 9) | r) << 9) | c;
}

// Bulk kernel: every block covers exactly 2048 pixels, no bounds checks,
// so no EXEC masking and no zero-init of the gather registers.
__global__ __launch_bounds__(256) void arnold7_gather_kernel(
    const float* __restrict__ in, float* __restrict__ out) {
  const unsigned base = blockIdx.x * (256u * PPT) + threadIdx.x;

  float v0[PPT], v1[PPT], v2[PPT];

  // Scattered gather burst: 8 independent b96 loads in flight per lane.
#pragma unroll
  for (int k = 0; k < PPT; ++k) {
    const float* s = in + 3u * src_index(base + (unsigned)k * 256u);
    v0[k] = s[0];
    v1[k] = s[1];
    v2[k] = s[2];
  }

  // Coalesced NT store burst: lane L writes 12 contiguous bytes at
  // out + 12*(base + k*256) -> 384 contiguous bytes per wave32 per k.
#pragma unroll
  for (int k = 0; k < PPT; ++k) {
    float* d = out + 3u * (base + (unsigned)k * 256u);
    __builtin_nontemporal_store(v0[k], d + 0);
    __builtin_nontemporal_store(v1[k], d + 1);
    __builtin_nontemporal_store(v2[k], d + 2);
  }
}

// Guarded tail for npix not divisible by 2048 (never taken for the harness
// shape 64*512*512; kept for shape robustness).
__global__ __launch_bounds__(256) void arnold7_tail_kernel(
    const float* __restrict__ in, float* __restrict__ out,
    unsigned start, unsigned npix) {
  const unsigned p = start + blockIdx.x * 256u + threadIdx.x;
  if (p < npix) {
    const float* s = in + 3u * src_index(p);
    float* d = out + 3u * p;
    __builtin_nontemporal_store(s[0], d + 0);
    __builtin_nontemporal_store(s[1], d + 1);
    __builtin_nontemporal_store(s[2], d + 2);
  }
}

extern "C" void kernel_launch(void* const* d_in, const int* in_sizes, int n_in,
                              void* d_out, int out_size, void* d_ws, size_t ws_size,
                              hipStream_t stream) {
  const float* in = (const float*)d_in[0];
  float* out = (float*)d_out;

  // in_sizes[0] = B*H*W*C with H=W=512, C=3.
  const unsigned npix = (unsigned)(in_sizes[0] / 3);
  const unsigned per_block = 256u * PPT;
  const unsigned full_blocks = npix / per_block;      // 8192 for the real shape
  const unsigned rem = npix - full_blocks * per_block; // 0 for the real shape

  if (full_blocks) {
    hipLaunchKernelGGL(arnold7_gather_kernel, dim3(full_blocks), dim3(256), 0,
                       stream, in, out);
  }
  if (rem) {
    const unsigned tail_blocks = (rem + 255u) / 256u;
    hipLaunchKernelGGL(arnold7_tail_kernel, dim3(tail_blocks), dim3(256), 0,
                       stream, in, out, full_blocks * per_block, npix);
  }
}